// PriorityQueueV0_57732950393175
// MI455X (gfx1250) — compile-verified
//
#include <hip/hip_runtime.h>
#include <hip/hip_bf16.h>

typedef float v2f __attribute__((ext_vector_type(2)));
typedef float v8f __attribute__((ext_vector_type(8)));

#if defined(__has_builtin)
#  if __has_builtin(__builtin_amdgcn_wmma_f32_16x16x4_f32)
#    define HAVE_WMMA_F32X4 1
#  endif
#endif

#define E_DIM 128
#define M_DIM 32
#define H_DIM 8

// ---------------------------------------------------------------------------
// K1: per-graph segment sum of z (batch is sorted -> contiguous ranges).
// One block per graph, 128 threads = one column each.
// ---------------------------------------------------------------------------
__global__ void k_segsum(const float* __restrict__ z, const int* __restrict__ batch,
                         float* __restrict__ Zsum, float* __restrict__ counts, int N) {
    int b = blockIdx.x;
    int e = threadIdx.x;
    int lo = 0, hi = N;
    while (lo < hi) { int mid = (lo + hi) >> 1; if (batch[mid] < b) lo = mid + 1; else hi = mid; }
    int start = lo;
    hi = N;
    while (lo < hi) { int mid = (lo + hi) >> 1; if (batch[mid] < b + 1) lo = mid + 1; else hi = mid; }
    int end = lo;
    float acc = 0.f;
    for (int n = start; n < end; ++n) acc += z[(size_t)n * E_DIM + e];
    Zsum[b * E_DIM + e] = acc;
    if (e == 0) counts[b] = (float)(end - start);
}

// ---------------------------------------------------------------------------
// K2: per-graph small work:
//   write_values = tanh(Zsum@Wv + cnt*bv)   (linearity of matmul vs segment_sum)
//   new_memory_values, new_write_mask (roll by 1), new_bias_mask
//   memv_att[b,m,h] = memory_values@Wa2 + ba2 ;  biasv = (bias_mask-1)*1e9
// ---------------------------------------------------------------------------
__global__ void k_graph_small(const float* __restrict__ memv,
                              const float* __restrict__ wmask,
                              const float* __restrict__ bmask,
                              const float* __restrict__ Wv, const float* __restrict__ bv,
                              const float* __restrict__ Wa2, const float* __restrict__ ba2,
                              const float* __restrict__ Zsum, const float* __restrict__ counts,
                              float* __restrict__ matt, float* __restrict__ biasv,
                              float* __restrict__ out_newmem,
                              float* __restrict__ out_nwm,
                              float* __restrict__ out_nbm) {
    int b = blockIdx.x;
    int e = threadIdx.x;
    float cnt = counts[b];
    float acc = bv[e] * cnt;
    #pragma unroll 8
    for (int k = 0; k < E_DIM; ++k) acc += Zsum[b * E_DIM + k] * Wv[k * E_DIM + e];
    float wval = tanhf(acc);
    for (int m = 0; m < M_DIM; ++m) {
        size_t idx = ((size_t)b * M_DIM + m) * E_DIM + e;
        out_newmem[idx] = memv[idx] + wval * wmask[b * M_DIM + m];
    }
    for (int p = e; p < M_DIM * H_DIM; p += E_DIM) {
        int m = p >> 3, h = p & 7;
        float a = ba2[h];
        const float* row = memv + ((size_t)b * M_DIM + m) * E_DIM;
        #pragma unroll 8
        for (int k = 0; k < E_DIM; ++k) a += row[k] * Wa2[k * H_DIM + h];
        matt[((size_t)b * M_DIM + m) * H_DIM + h] = a;
    }
    if (e < M_DIM) {
        int m = e;
        float bm = bmask[b * M_DIM + m];
        float wm = wmask[b * M_DIM + m];
        biasv[b * M_DIM + m] = (bm - 1.f) * 1e9f;
        out_nwm[b * M_DIM + m] = wmask[b * M_DIM + ((m + M_DIM - 1) & (M_DIM - 1))]; // roll shift=1
        out_nbm[b * M_DIM + m] = fminf(bm + wm, 1.f);
    }
}

// ---------------------------------------------------------------------------
// K2b: pack Wo into WMMA-B-fragment-friendly pairs:
//   WoP[p*E + c] = float2{ Wo[2p][c], Wo[2p+1][c] }   (p = 0..63)
// One b64 load per WMMA B fragment instead of two strided b32 loads.
// ---------------------------------------------------------------------------
__global__ void k_packWo(const float* __restrict__ Wo, float2* __restrict__ WoP) {
    int idx = blockIdx.x * blockDim.x + threadIdx.x;
    if (idx >= (E_DIM / 2) * E_DIM) return;
    int p = idx >> 7;          // row pair
    int c = idx & (E_DIM - 1); // column
    float2 v;
    v.x = Wo[(size_t)(2 * p) * E_DIM + c];
    v.y = Wo[(size_t)(2 * p + 1) * E_DIM + c];
    WoP[idx] = v;
}

// ---------------------------------------------------------------------------
// K3: att1 = z@Wa1 + ba1  (Wa1 staged in LDS), plus message_recepients.
// ---------------------------------------------------------------------------
__global__ void k_att1(const float* __restrict__ z, const float* __restrict__ Wa1,
                       const float* __restrict__ ba1,
                       float* __restrict__ att1, float* __restrict__ msg, int N) {
    __shared__ float sW[E_DIM * H_DIM];
    for (int i = threadIdx.x; i < E_DIM * H_DIM; i += blockDim.x) sW[i] = Wa1[i];
    __syncthreads();
    int n = blockIdx.x * blockDim.x + threadIdx.x;
    if (n >= N) return;
    float acc[H_DIM];
    #pragma unroll
    for (int h = 0; h < H_DIM; ++h) acc[h] = ba1[h];
    const float* zr = z + (size_t)n * E_DIM;
    #pragma unroll 4
    for (int k = 0; k < E_DIM; ++k) {
        float zv = zr[k];
        #pragma unroll
        for (int h = 0; h < H_DIM; ++h) acc[h] += zv * sW[k * H_DIM + h];
    }
    #pragma unroll
    for (int h = 0; h < H_DIM; ++h) att1[(size_t)n * H_DIM + h] = acc[h];
    msg[n] = (float)n;
}

// ---------------------------------------------------------------------------
// K4: main. One wave (32 lanes) per 16-node tile; lane == memory slot.
//  Phase1: logits -> leaky_relu -> +bias -> per-head softmax over slots
//          -> head-combine (Wh, bh) -> softmax over slots -> coefs in LDS
//  Phase2: agg[16x128] = coefs @ memory_values[batch] into LDS (float4 lanes)
//  Phase3: out = agg @ Wo + bo via v_wmma_f32_16x16x4_f32 (8 col-tiles x 32 K)
//          B fragments come from packed WoP (one b64 load each).
//          Wave-uniform fast path for the stores when the whole tile is valid.
// ---------------------------------------------------------------------------
__global__ void __launch_bounds__(128)
k_attention_out(const float* __restrict__ z, const int* __restrict__ batch,
                const float* __restrict__ memv,
                const float2* __restrict__ WoP, const float* __restrict__ bo,
                const float* __restrict__ Wh, const float* __restrict__ bh,
                const float* __restrict__ att1, const float* __restrict__ matt,
                const float* __restrict__ biasv,
                float* __restrict__ out, int N) {
    __shared__ float sAgg[4][16][E_DIM];   // 32 KB
    __shared__ float sCoef[4][16][M_DIM];  // 8 KB
    __shared__ int   sB[4][16];

    const int wave = threadIdx.x >> 5;
    const int lane = threadIdx.x & 31;
    const int tileBase = (blockIdx.x * 4 + wave) * 16;

    float wh[H_DIM];
    #pragma unroll
    for (int h = 0; h < H_DIM; ++h) wh[h] = Wh[h];
    const float bh0 = bh[0];

    // ---- Phase 1: coefficients (lane = slot m) ----
    for (int i = 0; i < 16; ++i) {
        int n = tileBase + i; if (n >= N) n = N - 1;
        int b = batch[n];
        float bias = biasv[b * M_DIM + lane];
        const float* mt = matt + ((size_t)b * M_DIM + lane) * H_DIM;
        const float* a1 = att1 + (size_t)n * H_DIM;
        float soft[H_DIM];
        #pragma unroll
        for (int h = 0; h < H_DIM; ++h) {
            float x = a1[h] + mt[h];
            x = (x > 0.f) ? x : 0.01f * x;          // leaky_relu
            soft[h] = x + bias;
        }
        #pragma unroll
        for (int h = 0; h < H_DIM; ++h) {
            float v = soft[h];
            float mx = v;
            #pragma unroll
            for (int off = 16; off >= 1; off >>= 1) mx = fmaxf(mx, __shfl_xor(mx, off));
            float ex = __expf(v - mx);
            float s = ex;
            #pragma unroll
            for (int off = 16; off >= 1; off >>= 1) s += __shfl_xor(s, off);
            soft[h] = ex / s;
        }
        float c2 = bh0;
        #pragma unroll
        for (int h = 0; h < H_DIM; ++h) c2 += soft[h] * wh[h];
        float mx2 = c2;
        #pragma unroll
        for (int off = 16; off >= 1; off >>= 1) mx2 = fmaxf(mx2, __shfl_xor(mx2, off));
        float e2 = __expf(c2 - mx2);
        float s2 = e2;
        #pragma unroll
        for (int off = 16; off >= 1; off >>= 1) s2 += __shfl_xor(s2, off);
        sCoef[wave][i][lane] = e2 / s2;
        if (lane == 0) sB[wave][i] = b;
    }
    __syncthreads();

    // ---- Phase 2: agg = coefs @ memory_values[b]  (lane spans E via float4) ----
    const float4* memv4 = (const float4*)memv;
    for (int i = 0; i < 16; ++i) {
        int b = sB[wave][i];
        const float4* row = memv4 + (size_t)b * M_DIM * (E_DIM / 4);
        const float* cf = &sCoef[wave][i][0];
        float ax = 0.f, ay = 0.f, az = 0.f, aw = 0.f;
        #pragma unroll 8
        for (int m = 0; m < M_DIM; ++m) {
            float c = cf[m];
            float4 v = row[m * (E_DIM / 4) + lane];
            ax += c * v.x; ay += c * v.y; az += c * v.z; aw += c * v.w;
        }
        float* dst = &sAgg[wave][i][lane * 4];
        dst[0] = ax; dst[1] = ay; dst[2] = az; dst[3] = aw;
    }
    __syncthreads();

    // ---- Phase 3: out_tile(16x128) = agg @ Wo + bo ----
    const int col = lane & 15;
    const bool hi = lane >= 16;
    const int koff = hi ? 2 : 0;   // A k-offset within a group of 4
    const int poff = hi ? 1 : 0;   // row-pair offset for packed B
    const bool fullTile = (tileBase + 16) <= N;

#if defined(HAVE_WMMA_F32X4)
    for (int j = 0; j < 8; ++j) {
        v8f acc = {};
        const int c = 16 * j + col;
        #pragma unroll 4
        for (int k = 0; k < 32; ++k) {
            const int kb = 4 * k + koff;
            const int p  = 2 * k + poff;
            v2f a, bf;
            a.x = sAgg[wave][col][kb];
            a.y = sAgg[wave][col][kb + 1];
            float2 w = WoP[(size_t)p * E_DIM + c];
            bf.x = w.x;
            bf.y = w.y;
            acc = __builtin_amdgcn_wmma_f32_16x16x4_f32(
                false, a, false, bf, (short)0, acc, false, false);
        }
        float bov = bo[c];
        float* outc = out + (size_t)(tileBase + (hi ? 8 : 0)) * E_DIM + c;
        if (fullTile) {
            // wave-uniform fast path: no exec-mask churn
            #pragma unroll
            for (int r = 0; r < 8; ++r) outc[(size_t)r * E_DIM] = acc[r] + bov;
        } else {
            #pragma unroll
            for (int r = 0; r < 8; ++r) {
                int nOut = tileBase + r + (hi ? 8 : 0);
                if (nOut < N) out[(size_t)nOut * E_DIM + c] = acc[r] + bov;
            }
        }
    }
#else
    // VALU fallback (should not be taken on gfx1250)
    for (int j = 0; j < 8; ++j) {
        const int c = 16 * j + col;
        float bov = bo[c];
        for (int r = 0; r < 8; ++r) {
            int rowN = r + (hi ? 8 : 0);
            int nOut = tileBase + rowN;
            float s = 0.f;
            for (int k = 0; k < E_DIM; ++k) {
                float2 w = WoP[(size_t)(k >> 1) * E_DIM + c];
                s += sAgg[wave][rowN][k] * ((k & 1) ? w.y : w.x);
            }
            if (nOut < N) out[(size_t)nOut * E_DIM + c] = s + bov;
        }
    }
#endif
}

// ---------------------------------------------------------------------------
extern "C" void kernel_launch(void* const* d_in, const int* in_sizes, int n_in,
                              void* d_out, int out_size, void* d_ws, size_t ws_size,
                              hipStream_t stream) {
    const float* z      = (const float*)d_in[0];
    const int*   batch  = (const int*)  d_in[1];
    const float* memv   = (const float*)d_in[2];
    const float* wmask  = (const float*)d_in[3];
    const float* bmask  = (const float*)d_in[4];
    const float* Wv     = (const float*)d_in[5];
    const float* bv     = (const float*)d_in[6];
    const float* Wo     = (const float*)d_in[7];
    const float* bo     = (const float*)d_in[8];
    const float* Wa1    = (const float*)d_in[9];
    const float* ba1    = (const float*)d_in[10];
    const float* Wa2    = (const float*)d_in[11];
    const float* ba2    = (const float*)d_in[12];
    const float* Wh     = (const float*)d_in[13];
    const float* bh     = (const float*)d_in[14];

    const int N = in_sizes[1];
    const int B = in_sizes[3] / M_DIM;

    // workspace layout (floats)
    float* ws     = (float*)d_ws;
    float* Zsum   = ws;                               // B*E
    float* counts = Zsum + (size_t)B * E_DIM;         // B
    float* matt   = counts + B;                       // B*M*H
    float* biasv  = matt + (size_t)B * M_DIM * H_DIM; // B*M
    float* att1   = biasv + (size_t)B * M_DIM;        // N*H
    float* WoPf   = att1 + (size_t)N * H_DIM;         // E*E (packed pairs)
    float2* WoP   = (float2*)WoPf;

    // output layout (floats, tuple order)
    float* out_output = (float*)d_out;
    float* out_msg    = out_output + (size_t)N * E_DIM;
    float* out_newmem = out_msg + N;
    float* out_nwm    = out_newmem + (size_t)B * M_DIM * E_DIM;
    float* out_nbm    = out_nwm + (size_t)B * M_DIM;

    k_segsum<<<B, E_DIM, 0, stream>>>(z, batch, Zsum, counts, N);
    k_graph_small<<<B, E_DIM, 0, stream>>>(memv, wmask, bmask, Wv, bv, Wa2, ba2,
                                           Zsum, counts, matt, biasv,
                                           out_newmem, out_nwm, out_nbm);
    k_packWo<<<((E_DIM / 2) * E_DIM + 255) / 256, 256, 0, stream>>>(Wo, WoP);
    k_att1<<<(N + 255) / 256, 256, 0, stream>>>(z, Wa1, ba1, att1, out_msg, N);

    int tiles = (N + 15) / 16;
    int blocks = (tiles + 3) / 4;
    k_attention_out<<<blocks, 128, 0, stream>>>(z, batch, memv, WoP, bo, Wh, bh,
                                                att1, matt, biasv, out_output, N);
}